// LabelLoss_15023795601462
// MI455X (gfx1250) — compile-verified
//
#include <hip/hip_runtime.h>

// ---------------------------------------------------------------------------
// LabelLoss for MI455X (gfx1250, wave32).
//   Stage 1: per-(b,j) heatmap argmax + loss  -> ws[B*J]   (bandwidth bound,
//            136 MB @ 23.3 TB/s ~ 5.8 us; fits in 192 MB L2 on replays)
//   Stage 2: out[b] = sum_j ws[b,j] via V_WMMA_F32_16X16X4_F32 (all-ones A,
//            layout-invariant total-sum trick), one wave per batch.
// ---------------------------------------------------------------------------

typedef __attribute__((ext_vector_type(2))) float v2f;
typedef __attribute__((ext_vector_type(8))) float v8f;

#define BB      32
#define JJ      17
#define MM      256
#define MN      (MM * MM)      // 65536 floats per (b,j) map
#define MN4     (MN / 4)       // float4 count
#define THREADS 256
#define WAVES   (THREADS / 32)

__global__ __launch_bounds__(THREADS)
void argmax_loss_kernel(const float* __restrict__ pred,
                        const float* __restrict__ gt,
                        const float* __restrict__ heatmap,
                        float* __restrict__ per_obj)
{
    const int bj  = blockIdx.x;          // 0 .. B*J-1
    const int tid = threadIdx.x;

    const float4* __restrict__ src =
        reinterpret_cast<const float4*>(heatmap + (size_t)bj * MN);

    // -------- thread-local argmax over 64 coalesced float4 loads ----------
    float bv = -__builtin_inff();
    int   bi = 0;
#pragma unroll 4
    for (int i = tid; i < MN4; i += THREADS) {
        float4 v = src[i];
        int base = i << 2;
        if (v.x > bv) { bv = v.x; bi = base;     }
        if (v.y > bv) { bv = v.y; bi = base + 1; }
        if (v.z > bv) { bv = v.z; bi = base + 2; }
        if (v.w > bv) { bv = v.w; bi = base + 3; }
    }

    // -------- wave32 butterfly reduction (first-max tie-break) ------------
    for (int off = 16; off > 0; off >>= 1) {
        float ov = __shfl_xor(bv, off, 32);
        int   oi = __shfl_xor(bi, off, 32);
        if (ov > bv || (ov == bv && oi < bi)) { bv = ov; bi = oi; }
    }

    // -------- cross-wave merge through LDS --------------------------------
    __shared__ float sval[WAVES];
    __shared__ int   sidx[WAVES];
    if ((tid & 31) == 0) { sval[tid >> 5] = bv; sidx[tid >> 5] = bi; }
    __syncthreads();

    if (tid == 0) {
        for (int w = 1; w < WAVES; ++w) {
            if (sval[w] > bv || (sval[w] == bv && sidx[w] < bi)) {
                bv = sval[w]; bi = sidx[w];
            }
        }
        // ---- per-object loss (a[x,y] == max value since m == n) ----------
        const float* p = pred + (size_t)bj * 11;
        const float* g = gt   + (size_t)bj * 11;

        float xf  = (float)(bi >> 8);        // idx // m
        float yf  = (float)(bi & (MM - 1));  // idx %  m
        float g9  = g[9];
        float g10 = g[10];
        bool valid = (g9 > 0.0f) && (g10 > 0.0f) &&
                     (g9 < (float)MM) && (g10 < (float)MM);

        float dx   = g9  + g[7] - xf - p[7];
        float dy   = g10 + g[8] - yf - p[8];
        float xy   = dx * dx + dy * dy;
        float om   = 1.0f - bv;
        float conf = om * om;
        float cls  = 0.0f;
#pragma unroll
        for (int k = 0; k < 7; ++k) { float d = p[k] - g[k]; cls += d * d; }

        per_obj[bj] = valid ? (cls + xy + conf) : 0.0f;
    }
}

// ---------------------------------------------------------------------------
// out[b] = sum_j per_obj[b,j].
// A = all-ones 16x4, C = 0  =>  D[m][n] = sum_k B[k][n] for every m, so the
// sum of D row 0 (documented layout: VGPR0, lanes 0-15) equals the sum of all
// 64 B slots regardless of the (undocumented) f32 B operand mapping.
// One full wave32 per batch; EXEC is all ones at the WMMA.
// ---------------------------------------------------------------------------
__global__ __launch_bounds__(32)
void batch_sum_wmma_kernel(const float* __restrict__ per_obj,
                           float* __restrict__ out)
{
    const int b    = blockIdx.x;
    const int lane = threadIdx.x;

    float v = (lane < JJ) ? per_obj[b * JJ + lane] : 0.0f;

    v2f Bm; Bm[0] = v;    Bm[1] = 0.0f;
    v2f Am; Am[0] = 1.0f; Am[1] = 1.0f;
    v8f c = {};

    v8f d = __builtin_amdgcn_wmma_f32_16x16x4_f32(
        /*neg_a=*/false, Am, /*neg_b=*/false, Bm,
        /*c_mod=*/(short)0, c, /*reuse_a=*/false, /*reuse_b=*/false);

    float s = d[0];            // row 0 (lanes 0-15); lanes 16-31 hold row 8 dup
    if (lane >= 16) s = 0.0f;  // drop the duplicate copy
    for (int off = 16; off > 0; off >>= 1) s += __shfl_xor(s, off, 32);

    if (lane == 0) out[b] = s;
}

extern "C" void kernel_launch(void* const* d_in, const int* in_sizes, int n_in,
                              void* d_out, int out_size, void* d_ws, size_t ws_size,
                              hipStream_t stream)
{
    const float* pred    = (const float*)d_in[0];  // [B, J, 11]
    const float* gt      = (const float*)d_in[1];  // [B, J, 11]
    const float* heatmap = (const float*)d_in[2];  // [B, J, 256, 256]
    float*       out     = (float*)d_out;          // [B]
    float*       per_obj = (float*)d_ws;           // [B*J] scratch (2176 B)

    argmax_loss_kernel<<<BB * JJ, THREADS, 0, stream>>>(pred, gt, heatmap, per_obj);
    batch_sum_wmma_kernel<<<BB, 32, 0, stream>>>(per_obj, out);
}